// HybridCoordinateStepper_20864951124544
// MI455X (gfx1250) — compile-verified
//
#include <hip/hip_runtime.h>
#include <math.h>
#include <stdint.h>

namespace {
constexpr int   KL    = 32;
constexpr int   NH    = 360;
constexpr int   NW    = 720;
constexpr int   SLICE = NH * NW;        // 259200 points per (b,k) slice
constexpr int   NCOL  = 2 * SLICE;      // 518400 columns (B=2); 2025 * 256 exactly
constexpr int   TPB   = 256;
constexpr float RGAS  = 287.0f;
constexpr float CPH   = 1004.0f;
constexpr float REA   = 6371000.0f;
constexpr float DAc   = 500.0f;         // DELTA_A (level-independent)
constexpr float DBc   = -1.0f / 32.0f;  // DELTA_B (level-independent)
constexpr double PI_D = 3.14159265358979323846;
}

// ---- gfx1250 async global->LDS path (guarded: falls back to direct loads) ----
#if defined(__HIP_DEVICE_COMPILE__) && defined(__gfx1250__) &&                 \
    __has_builtin(__builtin_amdgcn_global_load_async_to_lds_b64) &&            \
    __has_builtin(__builtin_amdgcn_global_load_async_to_lds_b32)
#define USE_ASYNC_LDS 1
#else
#define USE_ASYNC_LDS 0
#endif

#define GLOBAL_AS __attribute__((address_space(1)))
#define LDS_AS    __attribute__((address_space(3)))

#if USE_ASYNC_LDS
typedef int v2i __attribute__((vector_size(2 * sizeof(int))));

// generic->AS1: same 64-bit representation; generic->AS3: low 32 bits are the
// LDS offset (flat LDS aperture keeps the offset in addr[31:0]).
__device__ __forceinline__ GLOBAL_AS v2i* gptr64(const void* p) {
  return (GLOBAL_AS v2i*)(unsigned long long)(uintptr_t)p;
}
__device__ __forceinline__ LDS_AS v2i* lptr64(const void* p) {
  return (LDS_AS v2i*)(unsigned int)(uintptr_t)p;
}
__device__ __forceinline__ GLOBAL_AS int* gptr32(const void* p) {
  return (GLOBAL_AS int*)(unsigned long long)(uintptr_t)p;
}
__device__ __forceinline__ LDS_AS int* lptr32(const void* p) {
  return (LDS_AS int*)(unsigned int)(uintptr_t)p;
}
#endif

template <int N>
__device__ __forceinline__ void wait_async() {
#if USE_ASYNC_LDS
#if __has_builtin(__builtin_amdgcn_s_wait_asynccnt)
  __builtin_amdgcn_s_wait_asynccnt(N);
#else
  asm volatile("s_wait_asynccnt %0" ::"n"(N) : "memory");
#endif
#endif
}

__global__ __launch_bounds__(TPB) void hybrid_stepper(
    const float* __restrict__ uv_in,
    const float* __restrict__ T_in,
    const float* __restrict__ q_in,
    const float* __restrict__ ps_in,
    float2* __restrict__ duv_out,
    float*  __restrict__ dT_out,
    float*  __restrict__ dq_out,
    float*  __restrict__ dps_out)
{
  const int idx = blockIdx.x * blockDim.x + threadIdx.x;
  if (idx >= NCOL) return;
  const int tid = threadIdx.x;

  const int w = idx % NW;
  const int h = (idx / NW) % NH;
  const int b = idx / SLICE;

  // ---- geometry (once per column, double precision trig) ----
  const double dlat = PI_D / NH;
  const double dlon = 2.0 * PI_D / NW;
  const double latc = 0.5 * PI_D - (h + 0.5) * dlat;
  const int hm = (h > 0) ? h - 1 : 0;            // north neighbor row (clamped)
  const int hp = (h < NH - 1) ? h + 1 : h;       // south neighbor row (clamped)
  const float cos_c = (float)cos(latc);
  const float cos_n = (float)cos(0.5 * PI_D - (hm + 0.5) * dlat);
  const float cos_s = (float)cos(0.5 * PI_D - (hp + 0.5) * dlat);
  const float fcor  = (float)(2.0 * 7.292e-5 * sin(latc));
  const float inv2dlon   = (float)(1.0 / (2.0 * dlon));
  const float invm_dlat  = (float)(-1.0 / dlat);
  const float invm_2dlat = (float)(-1.0 / (2.0 * dlat));
  const float inv_recos  = 1.0f / (REA * cos_c);
  const float invRE      = 1.0f / REA;

  // d/dphi helper: xm = value at row h-1, xp = value at row h+1
  auto d_dphi = [&](float xm, float xc, float xp) -> float {
    if (h == 0)      return (xp - xc) * invm_dlat;   // (x[1]-x[0])/(-dlat)
    if (h == NH - 1) return (xc - xm) * invm_dlat;   // (x[N-1]-x[N-2])/(-dlat)
    return (xp - xm) * invm_2dlat;
  };

  // ---- stencil indices ----
  const int wm = (w == 0) ? NW - 1 : w - 1;
  const int wp = (w == NW - 1) ? 0 : w + 1;
  const int pc = h  * NW + w;
  const int pW = h  * NW + wm;
  const int pE = h  * NW + wp;
  const int pN = hm * NW + w;
  const int pS = hp * NW + w;
  const int psb = b * SLICE;

  const float ps_c = ps_in[psb + pc];
  const float ps_w = ps_in[psb + pW];
  const float ps_e = ps_in[psb + pE];
  const float ps_n = ps_in[psb + pN];
  const float ps_s = ps_in[psb + pS];

  const float gpse = (ps_e - ps_w) * inv2dlon * inv_recos;        // grad_s(p_s) east
  const float gpsn = d_dphi(ps_n, ps_c, ps_s) * invRE;            // grad_s(p_s) north
  const float dp_k = DAc + DBc * ps_c;                            // layer dp (same all k)

  const int b3 = b * KL * SLICE;
  const int iC = b3 + pc;
  const int iW = b3 + pW;
  const int iE = b3 + pE;
  const int iN = b3 + pN;
  const int iS = b3 + pS;
  const float2* __restrict__ uv2 = reinterpret_cast<const float2*>(uv_in);

  // =======================================================================
  // Pass 1: column-integrated continuity -> dp_s/dt
  // =======================================================================
  float dpsdt = 0.0f;
#pragma unroll 1
  for (int k = 0; k < KL; ++k) {
    const int o = k * SLICE;
    const float2 Uc = uv2[iC + o];
    const float2 Uw = uv2[iW + o];
    const float2 Ue = uv2[iE + o];
    const float2 Un = uv2[iN + o];
    const float2 Us = uv2[iS + o];
    const float div = ((Ue.x - Uw.x) * inv2dlon
                     + d_dphi(Un.y * cos_n, Uc.y * cos_c, Us.y * cos_s)) * inv_recos;
    const float vg  = Uc.x * gpse + Uc.y * gpsn;
    dpsdt += dp_k * div + DBc * vg;
    if (k + 6 < KL)  // k-stride ~2 MB: software prefetch into L2 (global_prefetch_b8)
      __builtin_prefetch((const void*)(uv2 + (iC + o + 6 * SLICE)), 0, 1);
  }
  dps_out[psb + pc] = dpsdt;

  // =======================================================================
  // Pass 2: streaming sweep surface->top with running accumulators.
  // Next-level center values are staged one iteration ahead with async
  // global->LDS copies (double buffered), retired with s_wait_asynccnt.
  // =======================================================================
#if USE_ASYNC_LDS
  __shared__ float2 s_uv[2][TPB];
  __shared__ float  s_T [2][TPB];
  __shared__ float  s_q [2][TPB];
  auto stage = [&](int kk) {  // kk uniform across the wave
    const int o = kk * SLICE;
    const int bs = kk & 1;
    __builtin_amdgcn_global_load_async_to_lds_b64(
        gptr64(uv2 + iC + o), lptr64(&s_uv[bs][tid]), 0, 0);
    __builtin_amdgcn_global_load_async_to_lds_b32(
        gptr32(T_in + iC + o), lptr32(&s_T[bs][tid]), 0, 0);
    __builtin_amdgcn_global_load_async_to_lds_b32(
        gptr32(q_in + iC + o), lptr32(&s_q[bs][tid]), 0, 0);
  };
#endif

  const float2 U0 = uv2[iC];
  float u_p = U0.x, u_c = U0.x, u_n = U0.x;
  float v_p = U0.y, v_c = U0.y, v_n = U0.y;
  float T_p, T_c, T_n;  T_p = T_c = T_n = T_in[iC];
  float q_p, q_c, q_n;  q_p = q_c = q_n = q_in[iC];
  float pm_p, pm_c, pm_n;
  pm_p = pm_c = pm_n = 250.0f + (1.0f - 1.0f / 64.0f) * ps_c;  // p_mid(k=0)

#if USE_ASYNC_LDS
  stage(1);
#endif

  float Chalf = 0.0f;                                   // C at interface k (surface = 0)
  float phiC = 0.f, phiW = 0.f, phiE = 0.f, phiN = 0.f, phiS = 0.f;  // hydrostatic phi
  float lpC = 0.f, lpW = 0.f, lpE = 0.f, lpN = 0.f, lpS = 0.f;       // log p at level k-1
  float TpC = 0.f, TpW = 0.f, TpE = 0.f, TpN = 0.f, TpS = 0.f;       // T at level k-1

#pragma unroll 1
  for (int k = 0; k < KL; ++k) {
    const int o = k * SLICE;
    const float amid = 500.0f * (float)k + 250.0f;
    const float bmid = 1.0f - (float)(2 * k + 1) * (1.0f / 64.0f);

    // mid-level pressure & its log at the 5 stencil columns
    const float pWk = amid + bmid * ps_w;
    const float pEk = amid + bmid * ps_e;
    const float pNk = amid + bmid * ps_n;
    const float pSk = amid + bmid * ps_s;
    const float lC = logf(pm_c);
    const float lW = logf(pWk);
    const float lE = logf(pEk);
    const float lN = logf(pNk);
    const float lS = logf(pSk);

    // hydrostatic: phi_k = phi_{k-1} + R*T_{k-1}*ln(p_{k-1}/p_k)
    if (k > 0) {
      phiC += RGAS * TpC * (lpC - lC);
      phiW += RGAS * TpW * (lpW - lW);
      phiE += RGAS * TpE * (lpE - lE);
      phiN += RGAS * TpN * (lpN - lN);
      phiS += RGAS * TpS * (lpS - lS);
    }

    // level-k neighbor loads (L1/L2-resident: stencil reuse across the wave)
    const float2 Uw = uv2[iW + o];
    const float2 Ue = uv2[iE + o];
    const float2 Un = uv2[iN + o];
    const float2 Us = uv2[iS + o];
    const float T_w  = T_in[iW + o], T_e  = T_in[iE + o];
    const float T_nn = T_in[iN + o], T_ss = T_in[iS + o];
    const float q_w  = q_in[iW + o], q_e  = q_in[iE + o];
    const float q_nn = q_in[iN + o], q_ss = q_in[iS + o];

    // center values at level k+1 (for dX/dp): async-LDS pipeline or direct
#if USE_ASYNC_LDS
    if (k + 2 < KL) stage(k + 2);          // issue 2 levels ahead
    if (k + 1 < KL) {
      if (k + 2 < KL) wait_async<1>(); else wait_async<0>();
      const int bs = (k + 1) & 1;
      const float2 Ux = s_uv[bs][tid];
      u_n = Ux.x;  v_n = Ux.y;
      T_n = s_T[bs][tid];
      q_n = s_q[bs][tid];
      pm_n = (amid + 500.0f) + (bmid - 1.0f / 32.0f) * ps_c;
    }
#else
    if (k < KL - 1) {
      const float2 Ux = uv2[iC + o + SLICE];
      u_n = Ux.x;  v_n = Ux.y;
      T_n = T_in[iC + o + SLICE];
      q_n = q_in[iC + o + SLICE];
      pm_n = (amid + 500.0f) + (bmid - 1.0f / 32.0f) * ps_c;
    }
#endif
    if (k + 6 < KL) {
      __builtin_prefetch((const void*)(T_in + (iC + o + 6 * SLICE)), 0, 1);
      __builtin_prefetch((const void*)(q_in + (iC + o + 6 * SLICE)), 0, 1);
    }

    // horizontal derivatives
    const float div  = ((Ue.x - Uw.x) * inv2dlon
                      + d_dphi(Un.y * cos_n, v_c * cos_c, Us.y * cos_s)) * inv_recos;
    const float zeta = ((Ue.y - Uw.y) * inv2dlon
                      - d_dphi(Un.x * cos_n, u_c * cos_c, Us.x * cos_s)) * inv_recos;
    const float vg = u_c * gpse + v_c * gpsn;

    // vertical mass-flux recurrence and omega
    const float term  = DBc * dpsdt + dp_k * div + DBc * vg;
    const float Cnext = Chalf - term;
    const float Cmid  = 0.5f * (Chalf + Cnext);
    Chalf = Cnext;
    const float omega = bmid * (dpsdt + vg) + Cmid;

    // grad_s(phi + ke)
    const float keC = 0.5f * (u_c * u_c + v_c * v_c);
    const float keW = 0.5f * (Uw.x * Uw.x + Uw.y * Uw.y);
    const float keE = 0.5f * (Ue.x * Ue.x + Ue.y * Ue.y);
    const float keN = 0.5f * (Un.x * Un.x + Un.y * Un.y);
    const float keS = 0.5f * (Us.x * Us.x + Us.y * Us.y);
    const float gpk_e = ((phiE + keE) - (phiW + keW)) * inv2dlon * inv_recos;
    const float gpk_n = d_dphi(phiN + keN, phiC + keC, phiS + keS) * invRE;

    // vertical FD in pressure (one-sided at ends)
    float du_dp, dv_dp, dT_dp, dq_dp;
    if (k == 0) {
      const float ip = 1.0f / (pm_n - pm_c);
      du_dp = (u_n - u_c) * ip;  dv_dp = (v_n - v_c) * ip;
      dT_dp = (T_n - T_c) * ip;  dq_dp = (q_n - q_c) * ip;
    } else if (k == KL - 1) {
      const float ip = 1.0f / (pm_c - pm_p);
      du_dp = (u_c - u_p) * ip;  dv_dp = (v_c - v_p) * ip;
      dT_dp = (T_c - T_p) * ip;  dq_dp = (q_c - q_p) * ip;
    } else {
      const float ip = 1.0f / (pm_n - pm_p);
      du_dp = (u_n - u_p) * ip;  dv_dp = (v_n - v_p) * ip;
      dT_dp = (T_n - T_p) * ip;  dq_dp = (q_n - q_p) * ip;
    }

    // tendencies
    const float avort = zeta + fcor;
    const float rtp   = RGAS * T_c / pm_c;
    const float dudt  =  avort * v_c - Cmid * du_dp - gpk_e - rtp * bmid * gpse;
    const float dvdt  = -avort * u_c - Cmid * dv_dp - gpk_n - rtp * bmid * gpsn;

    const float gT_e = (T_e - T_w) * inv2dlon * inv_recos;
    const float gT_n = d_dphi(T_nn, T_c, T_ss) * invRE;
    const float gq_e = (q_e - q_w) * inv2dlon * inv_recos;
    const float gq_n = d_dphi(q_nn, q_c, q_ss) * invRE;
    const float dTdt = -(u_c * gT_e + v_c * gT_n) - Cmid * dT_dp
                       + RGAS * T_c * omega / (CPH * pm_c);
    const float dqdt = -(u_c * gq_e + v_c * gq_n) - Cmid * dq_dp;

    duv_out[iC + o] = make_float2(dudt, dvdt);
    dT_out[iC + o]  = dTdt;
    dq_out[iC + o]  = dqdt;

    // rotate running state
    TpC = T_c; TpW = T_w; TpE = T_e; TpN = T_nn; TpS = T_ss;
    lpC = lC;  lpW = lW;  lpE = lE;  lpN = lN;   lpS = lS;
    u_p = u_c; u_c = u_n;
    v_p = v_c; v_c = v_n;
    T_p = T_c; T_c = T_n;
    q_p = q_c; q_c = q_n;
    pm_p = pm_c; pm_c = pm_n;
  }
}

extern "C" void kernel_launch(void* const* d_in, const int* in_sizes, int n_in,
                              void* d_out, int out_size, void* d_ws, size_t ws_size,
                              hipStream_t stream) {
  (void)in_sizes; (void)n_in; (void)d_ws; (void)ws_size; (void)out_size;
  const float* uv = (const float*)d_in[0];
  const float* T  = (const float*)d_in[1];
  const float* q  = (const float*)d_in[2];
  const float* ps = (const float*)d_in[3];

  float* out = (float*)d_out;
  float2* duv = (float2*)out;                                   // 2*32*360*720*2 floats
  float*  dT  = out + (size_t)2 * KL * SLICE * 2;               // +33,177,600
  float*  dq  = dT  + (size_t)2 * KL * SLICE;                   // +16,588,800
  float*  dps = dq  + (size_t)2 * KL * SLICE;                   // +16,588,800

  const int threads = TPB;
  const int blocks  = (NCOL + threads - 1) / threads;           // 2025 (exact)
  hybrid_stepper<<<blocks, threads, 0, stream>>>(uv, T, q, ps, duv, dT, dq, dps);
}